// Distance_20959440404484
// MI455X (gfx1250) — compile-verified
//
#include <hip/hip_runtime.h>

// Pairwise squared Euclidean distance:
//   dist[i,j] = ||s_i||^2 + ||t_j||^2 - 2 * (s @ t^T)[i,j]
// Core: f32 GEMM M=N=8192, K=512 on V_WMMA_F32_16X16X4_F32 (exact f32 math).
// Compute-bound on the WMMA pipe; inputs L2-resident (32 MB << 192 MB L2).
// Output (256 MB > L2) is streamed with non-temporal stores so it does not
// evict the re-read inputs.

typedef __attribute__((ext_vector_type(2))) float v2f;
typedef __attribute__((ext_vector_type(8))) float v8f;

#define M_DIM 8192
#define N_DIM 8192
#define K_DIM 512

#define BM 128           // block tile M
#define BN 128           // block tile N
#define KB 32            // K chunk staged in LDS
#define LDST 36          // LDS row stride (floats): 16B-aligned, conflict-avoiding
#define NCHUNK (K_DIM / KB)

// ---------------------------------------------------------------------------
// Row squared-norm pre-pass: one wave32 per row, float4 coalesced loads,
// shuffle reduction. 8192 rows x 2 KB each.
// ---------------------------------------------------------------------------
__global__ __launch_bounds__(256) void row_sq_norms(const float* __restrict__ x,
                                                    float* __restrict__ o) {
    const int wave = blockIdx.x * 8 + (threadIdx.x >> 5);
    const int lane = threadIdx.x & 31;
    const float4* row = (const float4*)(x + (size_t)wave * K_DIM);
    float acc = 0.f;
#pragma unroll
    for (int i = 0; i < 4; ++i) {               // 128 float4 per row / 32 lanes
        float4 v = row[lane + 32 * i];
        acc += v.x * v.x + v.y * v.y + v.z * v.z + v.w * v.w;
    }
#pragma unroll
    for (int off = 16; off > 0; off >>= 1)
        acc += __shfl_xor(acc, off, 32);        // wave32 reduction
    if (lane == 0) o[wave] = acc;
}

// ---------------------------------------------------------------------------
// Fused distance GEMM.
// Block: 256 threads = 8 waves (2 along M x 4 along N), tile 128x128.
// Each wave: 64x32 patch = 4x2 WMMA 16x16 f32 accumulators (8 chains of ILP,
// 6 fragment loads per 8 WMMAs).
// ---------------------------------------------------------------------------
__global__ __launch_bounds__(256) void dist_wmma(const float* __restrict__ S,
                                                 const float* __restrict__ T,
                                                 const float* __restrict__ ssq,
                                                 const float* __restrict__ tsq,
                                                 float* __restrict__ out) {
    __shared__ float As[BM * LDST];   // 128*36*4 = 18 KB
    __shared__ float Bs[BN * LDST];   // 128*36*4 = 18 KB

    const int tid  = threadIdx.x;
    const int lane = tid & 31;
    const int wave = tid >> 5;
    const int l16  = lane & 15;
    const int half = lane >> 4;                 // lane half selects K pair / M+8

    const int mBase = blockIdx.y * BM;
    const int nBase = blockIdx.x * BN;
    const int waveM = (wave & 1) * 64;          // 2 waves along M, 64 rows each
    const int waveN = (wave >> 1) * 32;         // 4 waves along N, 32 cols each

    const float* Sg = S + (size_t)mBase * K_DIM;
    const float* Tg = T + (size_t)nBase * K_DIM;

    v8f acc[4][2] = {};                         // 8 x (16x16 f32) accumulators

    // ---- load K-chunk 0 into registers then LDS ----
    // A/B tiles: 128 rows x 8 float4 = 1024 float4 each; 256 threads x 4.
    float4 pa[4], pb[4];
#pragma unroll
    for (int i = 0; i < 4; ++i) {
        int idx = tid + 256 * i;
        int r = idx >> 3, c = (idx & 7) * 4;
        pa[i] = *(const float4*)(Sg + (size_t)r * K_DIM + c);
        pb[i] = *(const float4*)(Tg + (size_t)r * K_DIM + c);
    }
#pragma unroll
    for (int i = 0; i < 4; ++i) {
        int idx = tid + 256 * i;
        *(float4*)&As[(idx >> 3) * LDST + (idx & 7) * 4] = pa[i];
        *(float4*)&Bs[(idx >> 3) * LDST + (idx & 7) * 4] = pb[i];
    }
    __syncthreads();

    for (int kc = 0; kc < NCHUNK; ++kc) {
        // ---- register-prefetch next chunk (overlaps with WMMA below) ----
        if (kc + 1 < NCHUNK) {
            const int k0 = (kc + 1) * KB;
#pragma unroll
            for (int i = 0; i < 4; ++i) {
                int idx = tid + 256 * i;
                int r = idx >> 3, c = (idx & 7) * 4;
                pa[i] = *(const float4*)(Sg + (size_t)r * K_DIM + k0 + c);
                pb[i] = *(const float4*)(Tg + (size_t)r * K_DIM + k0 + c);
            }
        }

        // ---- 8 WMMA k-steps over the chunk ----
        // A 16x4 f32 layout: lane half picks K {0,1} vs {2,3}; v0/v1 = K even/odd.
        // Per lane: 2 consecutive floats at column k4*4 + 2*half (8B-aligned).
#pragma unroll
        for (int k4 = 0; k4 < KB / 4; ++k4) {
            const int kcol = k4 * 4 + 2 * half;
            v2f a[4], b[2];
#pragma unroll
            for (int mt = 0; mt < 4; ++mt)
                a[mt] = *(const v2f*)&As[(waveM + mt * 16 + l16) * LDST + kcol];
#pragma unroll
            for (int nt = 0; nt < 2; ++nt)
                b[nt] = *(const v2f*)&Bs[(waveN + nt * 16 + l16) * LDST + kcol];

#pragma unroll
            for (int mt = 0; mt < 4; ++mt)
#pragma unroll
                for (int nt = 0; nt < 2; ++nt)
                    acc[mt][nt] = __builtin_amdgcn_wmma_f32_16x16x4_f32(
                        false, a[mt], false, b[nt], (short)0, acc[mt][nt],
                        false, false);
        }
        __syncthreads();

        // ---- commit prefetched chunk to LDS ----
        if (kc + 1 < NCHUNK) {
#pragma unroll
            for (int i = 0; i < 4; ++i) {
                int idx = tid + 256 * i;
                *(float4*)&As[(idx >> 3) * LDST + (idx & 7) * 4] = pa[i];
                *(float4*)&Bs[(idx >> 3) * LDST + (idx & 7) * 4] = pb[i];
            }
            __syncthreads();
        }
    }

    // ---- epilogue: dist = ||s||^2 + ||t||^2 - 2*cross ----
    // C/D 16x16 layout: lane l16 -> N, half -> M+8, VGPR v -> M+v.
    // Non-temporal stores: 256 MB write-once output must not evict the
    // L2-resident inputs.
#pragma unroll
    for (int mt = 0; mt < 4; ++mt) {
        const int m0 = mBase + waveM + mt * 16 + half * 8;   // multiple of 8
        float4 sq0 = *(const float4*)&ssq[m0];
        float4 sq1 = *(const float4*)&ssq[m0 + 4];
        float sm[8] = {sq0.x, sq0.y, sq0.z, sq0.w, sq1.x, sq1.y, sq1.z, sq1.w};
#pragma unroll
        for (int nt = 0; nt < 2; ++nt) {
            const int n = nBase + waveN + nt * 16 + l16;
            const float tq = tsq[n];
#pragma unroll
            for (int v = 0; v < 8; ++v) {
                float d = sm[v] + tq - 2.0f * acc[mt][nt][v];
                __builtin_nontemporal_store(d, &out[(size_t)(m0 + v) * N_DIM + n]);
            }
        }
    }
}

// ---------------------------------------------------------------------------
extern "C" void kernel_launch(void* const* d_in, const int* in_sizes, int n_in,
                              void* d_out, int out_size, void* d_ws, size_t ws_size,
                              hipStream_t stream) {
    const float* s = (const float*)d_in[0];
    const float* t = (const float*)d_in[1];
    float* out = (float*)d_out;

    float* ssq = (float*)d_ws;          // [8192]
    float* tsq = ssq + M_DIM;           // [8192]  (64 KB total scratch)

    row_sq_norms<<<M_DIM / 8, 256, 0, stream>>>(s, ssq);
    row_sq_norms<<<N_DIM / 8, 256, 0, stream>>>(t, tsq);

    dim3 grid(N_DIM / BN, M_DIM / BM);  // 64 x 64 = 4096 workgroups
    dist_wmma<<<grid, 256, 0, stream>>>(s, t, ssq, tsq, out);
}